// Seq2Seq_75677323755848
// MI455X (gfx1250) — compile-verified
//
#include <hip/hip_runtime.h>
#include <math.h>

// ---------------------------------------------------------------------------
// Problem constants (from reference setup_inputs)
// ---------------------------------------------------------------------------
#define BB   132
#define SS   50
#define TT   50
#define EE   128
#define HH   256
#define HD   256
#define AA   32
#define VV   10000

#define KS_BIG 4          // split-K factor for the dominant 132x10000x896 GEMM
#define KC_BIG 224        // 896 / 4, multiple of 4

typedef float v2f __attribute__((ext_vector_type(2)));
typedef float v8f __attribute__((ext_vector_type(8)));

// ---------------------------------------------------------------------------
// Generic fp32 WMMA GEMM:  D[m,n] = act( sum_k X[m*ldx+k] * W[n*ldw+k] + bias[n] )
//   X: (M,K) rows stride ldx ; W: (N,K) rows stride ldw (torch-style weights)
//   D: rows stride ldd (64-bit so we can write strided into d_out)
// Wave tile: (16*SUB) x 64  -> SUB*4 x V_WMMA_F32_16X16X4_F32 per K-step.
// fp32 A tile (16x4): lane (g=lane>>4, m=lane&15) holds X[m][k+2g+0..1]
// fp32 B tile (4x16): lane (g, n=lane&15)         holds W[n][k+2g+0..1]
// C/D tile:           vgpr r, lane -> row = r + 8*(lane>>4), col = lane&15
// Software pipelined: next K-group's A/B loads issue before current WMMAs.
// Split-K: KS chunks of Kc; KS>1 writes partials to D[ks*M*N + m*N + n]
// (bias then applied by reduce_splitk kernel, which applies no activation).
// ---------------------------------------------------------------------------
template<int ACT, int SUB>  // ACT: 0 none, 1 tanh ; SUB: A-subtiles (rows = 16*SUB)
__global__ void gemm_wmma_f32(const float* __restrict__ X, int ldx,
                              const float* __restrict__ W, int ldw,
                              const float* __restrict__ bias,
                              float* __restrict__ D, long long ldd,
                              int M, int N, int K,
                              int KS, int Kc)
{
    const int lane  = threadIdx.x & 31;
    const int wave  = threadIdx.x >> 5;
    const int wpb   = blockDim.x >> 5;
    const int gwave = blockIdx.x * wpb + wave;
    const int nwave = gridDim.x * wpb;

    const int MR      = 16 * SUB;
    const int tilesN  = (N + 63) >> 6;
    const int tilesM  = (M + MR - 1) / MR;
    const int tilesMN = tilesM * tilesN;
    const int tiles   = tilesMN * KS;

    const int g = lane >> 4;   // K-pair selector
    const int l = lane & 15;   // A-row / B-col within tile

    for (int t = gwave; t < tiles; t += nwave) {
        const int ks = t / tilesMN;
        const int mn = t - ks * tilesMN;
        const int mt = mn / tilesN;
        const int nt = mn - mt * tilesN;
        const int m0 = mt * MR;
        const int n0 = nt << 6;

        const int kbeg = ks * Kc;
        int kend = kbeg + Kc; if (kend > K) kend = K;

        const float* xr[SUB];
        #pragma unroll
        for (int u = 0; u < SUB; ++u) {
            int r = m0 + 16 * u + l; if (r >= M) r = M - 1;
            xr[u] = X + (size_t)r * ldx + 2 * g;
        }
        const float* wr[4];
        #pragma unroll
        for (int j = 0; j < 4; ++j) {
            int c = n0 + (j << 4) + l; if (c >= N) c = N - 1;
            wr[j] = W + (size_t)c * ldw + 2 * g;
        }

        v8f acc[SUB * 4];
        #pragma unroll
        for (int q = 0; q < SUB * 4; ++q) acc[q] = (v8f){};

        // --- software-pipelined K loop (all K chunks here are >= 8) ---
        v2f a[SUB], b[4];
        #pragma unroll
        for (int u = 0; u < SUB; ++u) a[u] = *(const v2f*)(xr[u] + kbeg);
        #pragma unroll
        for (int j = 0; j < 4; ++j)  b[j] = *(const v2f*)(wr[j] + kbeg);

        int k = kbeg;
        for (; k + 4 < kend; k += 4) {
            v2f an[SUB], bn[4];
            #pragma unroll
            for (int u = 0; u < SUB; ++u) an[u] = *(const v2f*)(xr[u] + k + 4);
            #pragma unroll
            for (int j = 0; j < 4; ++j)  bn[j] = *(const v2f*)(wr[j] + k + 4);
            #pragma unroll
            for (int u = 0; u < SUB; ++u)
                #pragma unroll
                for (int j = 0; j < 4; ++j)
                    acc[u * 4 + j] = __builtin_amdgcn_wmma_f32_16x16x4_f32(
                        false, a[u], false, b[j], (short)0, acc[u * 4 + j], false, false);
            #pragma unroll
            for (int u = 0; u < SUB; ++u) a[u] = an[u];
            #pragma unroll
            for (int j = 0; j < 4; ++j)  b[j] = bn[j];
        }
        #pragma unroll
        for (int u = 0; u < SUB; ++u)
            #pragma unroll
            for (int j = 0; j < 4; ++j)
                acc[u * 4 + j] = __builtin_amdgcn_wmma_f32_16x16x4_f32(
                    false, a[u], false, b[j], (short)0, acc[u * 4 + j], false, false);

        // --- store ---
        float* Dp = D;
        long long ld = ldd;
        if (KS > 1) { Dp = D + (size_t)ks * M * N; ld = N; }

        #pragma unroll
        for (int j = 0; j < 4; ++j) {
            const int col = n0 + (j << 4) + l;
            if (col < N) {
                const float bv = (KS == 1 && bias) ? bias[col] : 0.0f;
                #pragma unroll
                for (int u = 0; u < SUB; ++u) {
                    #pragma unroll
                    for (int r = 0; r < 8; ++r) {
                        const int row = m0 + 16 * u + r + (g << 3);
                        if (row < M) {
                            float v = acc[u * 4 + j][r] + bv;
                            if (ACT == 1) v = tanhf(v);
                            Dp[(size_t)row * ld + col] = v;
                        }
                    }
                }
            }
        }
    }
}

// sum split-K partials + bias -> strided destination
__global__ void reduce_splitk(const float* __restrict__ part,  // (KS, M, N)
                              const float* __restrict__ bias,
                              float* __restrict__ D, long long ldd,
                              int M, int N, int KS) {
    int i = blockIdx.x * blockDim.x + threadIdx.x;
    if (i >= M * N) return;
    int m = i / N, n = i - m * N;
    float acc = bias ? bias[n] : 0.0f;
    for (int ks = 0; ks < KS; ++ks) acc += part[(size_t)ks * M * N + i];
    D[(size_t)m * ldd + n] = acc;
}

// ---------------------------------------------------------------------------
// Elementwise helpers
// ---------------------------------------------------------------------------
__global__ void zero_f32(float* __restrict__ p, int n) {
    int i = blockIdx.x * blockDim.x + threadIdx.x;
    if (i < n) p[i] = 0.0f;
}

__global__ void zero_out_slice0(float* __restrict__ out) {
    int i = blockIdx.x * blockDim.x + threadIdx.x;
    if (i >= BB * VV) return;
    int b = i / VV, v = i - b * VV;
    out[(size_t)b * TT * VV + v] = 0.0f;
}

__global__ void enc_embed(const int* __restrict__ src, const float* __restrict__ emb,
                          float* __restrict__ x) {
    int i = blockIdx.x * blockDim.x + threadIdx.x;
    if (i >= BB * SS * EE) return;
    int r = i / EE, j = i - r * EE;
    x[i] = emb[(size_t)src[r] * EE + j];
}

// GRU gates; optionally mirrors h_new into out2[b*out2_stride + n] (decoder outcat)
__global__ void gru_elem(const float* __restrict__ xp, long long xp_stride,
                         const float* __restrict__ gh,
                         const float* __restrict__ hprev,
                         float* __restrict__ hnew,
                         float* __restrict__ out2, long long out2_stride,
                         int B, int H) {
    int i = blockIdx.x * blockDim.x + threadIdx.x;
    if (i >= B * H) return;
    int b = i / H, n = i - b * H;
    const float* xr = xp + (size_t)b * xp_stride;
    const float* gr = gh + (size_t)b * 3 * H;
    float r  = 1.0f / (1.0f + __expf(-(xr[n]     + gr[n])));
    float z  = 1.0f / (1.0f + __expf(-(xr[n + H] + gr[n + H])));
    float nn = tanhf(xr[n + 2 * H] + r * gr[n + 2 * H]);
    float v  = (1.0f - z) * nn + z * hprev[i];
    hnew[i] = v;
    if (out2) out2[(size_t)b * out2_stride + n] = v;
}

__global__ void pack_enc_out(const float* __restrict__ hf_seq,
                             const float* __restrict__ hb_seq,
                             float* __restrict__ enc_out) {
    int i = blockIdx.x * blockDim.x + threadIdx.x;
    if (i >= BB * SS * 2 * HH) return;
    int j  = i & (2 * HH - 1);
    int bs = i / (2 * HH);
    int b  = bs / SS, s = bs - b * SS;
    float v = (j < HH) ? hf_seq[((size_t)s * BB + b) * HH + j]
                       : hb_seq[((size_t)s * BB + b) * HH + (j - HH)];
    enc_out[i] = v;
}

__global__ void enc_cat_final(const float* __restrict__ hf_seq,
                              const float* __restrict__ hb_seq,
                              float* __restrict__ cat) {
    int i = blockIdx.x * blockDim.x + threadIdx.x;
    if (i >= BB * 2 * HH) return;
    int b = i / (2 * HH), j = i - b * (2 * HH);
    cat[i] = (j < HH) ? hf_seq[((size_t)(SS - 1) * BB + b) * HH + j]
                      : hb_seq[(size_t)b * HH + (j - HH)];
}

// ---------------------------------------------------------------------------
// Fused decoder attention step: one block per batch row, 256 threads.
//   hp[a]    = h[b] . wa_dec[a]                 (a = lane, A == 32 == wave32)
//   e[s]     = sum_a tanh(hp[a] + enc_proj[b,s,a])
//   attn     = softmax(e)
//   w        = sum_s attn[s] * enc_out[b,s,:]
// Writes xcat[b]   = [emb_t, w]            (128 + 512)
//        outcat[b] = [  - , w, emb_t]      ([256:768], [768:896]; [0:256] by GRU)
// ---------------------------------------------------------------------------
__global__ void attn_fused(const float* __restrict__ h,         // (B, HD)
                           const float* __restrict__ attn_w,    // (A, HD+2H), wa_dec = [:, :HD]
                           const float* __restrict__ enc_proj,  // (B, S, A)
                           const float* __restrict__ enc_out,   // (B, S, 2H)
                           const int* __restrict__ trg, int t,
                           const float* __restrict__ dec_emb,   // (V, E)
                           float* __restrict__ xcat,            // (B, E+2H)
                           float* __restrict__ outcat)          // (B, HD+2H+E)
{
    __shared__ float sE[SS];
    const int b    = blockIdx.x;
    const int tid  = threadIdx.x;
    const int lane = tid & 31;
    const int wv   = tid >> 5;

    // per-lane attention-decoder projection (duplicated across waves; L1-resident)
    float hp = 0.0f;
    const float* hrow = h + b * HD;
    const float* wrow = attn_w + (size_t)lane * (HD + 2 * HH);
    for (int k = 0; k < HD; ++k) hp += hrow[k] * wrow[k];

    for (int s = wv; s < SS; s += 8) {
        float v = tanhf(hp + enc_proj[((size_t)b * SS + s) * AA + lane]);
        #pragma unroll
        for (int off = 16; off > 0; off >>= 1) v += __shfl_xor(v, off, 32);
        if (lane == 0) sE[s] = v;
    }
    __syncthreads();

    if (tid == 0) {
        float mx = sE[0];
        for (int s = 1; s < SS; ++s) mx = fmaxf(mx, sE[s]);
        float sum = 0.0f;
        for (int s = 0; s < SS; ++s) { float e = __expf(sE[s] - mx); sE[s] = e; sum += e; }
        float inv = 1.0f / sum;
        for (int s = 0; s < SS; ++s) sE[s] *= inv;
    }
    __syncthreads();

    // weighted encoder sum -> both concat buffers
    for (int j = tid; j < 2 * HH; j += blockDim.x) {
        float acc = 0.0f;
        const float* eo = enc_out + (size_t)b * SS * 2 * HH + j;
        for (int s = 0; s < SS; ++s) acc += sE[s] * eo[(size_t)s * 2 * HH];
        xcat[b * (EE + 2 * HH) + EE + j]        = acc;
        outcat[b * (HD + 2 * HH + EE) + HD + j] = acc;
    }
    // embedding gather -> both concat buffers
    const int tok = trg[b * TT + t];
    for (int j = tid; j < EE; j += blockDim.x) {
        float e = dec_emb[(size_t)tok * EE + j];
        xcat[b * (EE + 2 * HH) + j]                      = e;
        outcat[b * (HD + 2 * HH + EE) + HD + 2 * HH + j] = e;
    }
}

// ---------------------------------------------------------------------------
// Host side
// ---------------------------------------------------------------------------
static inline int cdiv(int a, int b) { return (a + b - 1) / b; }
static inline int gemm_grid(int M, int N, int mrows, int ks) {
    int tiles = cdiv(M, mrows) * cdiv(N, 64) * ks;
    return cdiv(tiles, 8);   // 8 waves per 256-thread block
}

extern "C" void kernel_launch(void* const* d_in, const int* in_sizes, int n_in,
                              void* d_out, int out_size, void* d_ws, size_t ws_size,
                              hipStream_t stream) {
    (void)in_sizes; (void)n_in; (void)out_size; (void)ws_size;

    const int*   src       = (const int*)  d_in[0];
    const int*   trg       = (const int*)  d_in[1];
    const float* enc_emb   = (const float*)d_in[2];
    const float* enc_wih_f = (const float*)d_in[3];
    const float* enc_whh_f = (const float*)d_in[4];
    const float* enc_bih_f = (const float*)d_in[5];
    const float* enc_bhh_f = (const float*)d_in[6];
    const float* enc_wih_b = (const float*)d_in[7];
    const float* enc_whh_b = (const float*)d_in[8];
    const float* enc_bih_b = (const float*)d_in[9];
    const float* enc_bhh_b = (const float*)d_in[10];
    const float* enc_fc_w  = (const float*)d_in[11];
    const float* enc_fc_b  = (const float*)d_in[12];
    const float* attn_w    = (const float*)d_in[13];   // (32, 768): [:, :256]=wa_dec, [:,256:]=wa_enc
    const float* attn_b    = (const float*)d_in[14];
    const float* dec_emb   = (const float*)d_in[15];
    const float* dec_wih   = (const float*)d_in[16];
    const float* dec_whh   = (const float*)d_in[17];
    const float* dec_bih   = (const float*)d_in[18];
    const float* dec_bhh   = (const float*)d_in[19];
    const float* dec_fc_w  = (const float*)d_in[20];
    const float* dec_fc_b  = (const float*)d_in[21];
    float* out = (float*)d_out;

    // ---- workspace carve-up (floats) ----
    float* ws = (float*)d_ws;
    size_t off = 0;
    float* x_emb    = ws + off; off += (size_t)BB * SS * EE;
    float* xp_f     = ws + off; off += (size_t)BB * SS * 3 * HH;
    float* xp_b     = ws + off; off += (size_t)BB * SS * 3 * HH;
    float* hf_seq   = ws + off; off += (size_t)SS * BB * HH;        // (s,b,H)
    float* hb_seq   = ws + off; off += (size_t)SS * BB * HH;        // (s,b,H) original time order
    float* enc_out  = ws + off; off += (size_t)BB * SS * 2 * HH;
    float* enc_proj = ws + off; off += (size_t)BB * SS * AA;
    float* zero_h   = ws + off; off += (size_t)BB * HH;
    float* hbuf0    = ws + off; off += (size_t)BB * HD;
    float* hbuf1    = ws + off; off += (size_t)BB * HD;
    float* gh_buf   = ws + off; off += (size_t)BB * 3 * HH;
    float* henc_cat = ws + off; off += (size_t)BB * 2 * HH;
    float* xcat     = ws + off; off += (size_t)BB * (EE + 2 * HH);
    float* xp_dec   = ws + off; off += (size_t)BB * 3 * HD;
    float* outcat   = ws + off; off += (size_t)BB * (HD + 2 * HH + EE);
    float* part     = ws + off; off += (size_t)KS_BIG * BB * VV;    // split-K partials

    const int EW = 256;

    // ---- encoder ----
    zero_f32<<<cdiv(BB * HH, EW), EW, 0, stream>>>(zero_h, BB * HH);
    zero_out_slice0<<<cdiv(BB * VV, EW), EW, 0, stream>>>(out);

    enc_embed<<<cdiv(BB * SS * EE, EW), EW, 0, stream>>>(src, enc_emb, x_emb);

    // xp_f / xp_b : (B*S, 3H) = x_emb (B*S, E) @ wih^T + bih   (wide M -> 32-row tiles)
    gemm_wmma_f32<0, 2><<<gemm_grid(BB * SS, 3 * HH, 32, 1), 256, 0, stream>>>(
        x_emb, EE, enc_wih_f, EE, enc_bih_f, xp_f, 3 * HH, BB * SS, 3 * HH, EE, 1, EE);
    gemm_wmma_f32<0, 2><<<gemm_grid(BB * SS, 3 * HH, 32, 1), 256, 0, stream>>>(
        x_emb, EE, enc_wih_b, EE, enc_bih_b, xp_b, 3 * HH, BB * SS, 3 * HH, EE, 1, EE);

    // forward scan
    for (int s = 0; s < SS; ++s) {
        const float* hprev = (s == 0) ? zero_h : hf_seq + (size_t)(s - 1) * BB * HH;
        gemm_wmma_f32<0, 1><<<gemm_grid(BB, 3 * HH, 16, 1), 256, 0, stream>>>(
            hprev, HH, enc_whh_f, HH, enc_bhh_f, gh_buf, 3 * HH, BB, 3 * HH, HH, 1, HH);
        gru_elem<<<cdiv(BB * HH, EW), EW, 0, stream>>>(
            xp_f + (size_t)s * 3 * HH, (long long)SS * 3 * HH,
            gh_buf, hprev, hf_seq + (size_t)s * BB * HH, nullptr, 0, BB, HH);
    }
    // backward scan (original time order; step s uses h from s+1)
    for (int s = SS - 1; s >= 0; --s) {
        const float* hprev = (s == SS - 1) ? zero_h : hb_seq + (size_t)(s + 1) * BB * HH;
        gemm_wmma_f32<0, 1><<<gemm_grid(BB, 3 * HH, 16, 1), 256, 0, stream>>>(
            hprev, HH, enc_whh_b, HH, enc_bhh_b, gh_buf, 3 * HH, BB, 3 * HH, HH, 1, HH);
        gru_elem<<<cdiv(BB * HH, EW), EW, 0, stream>>>(
            xp_b + (size_t)s * 3 * HH, (long long)SS * 3 * HH,
            gh_buf, hprev, hb_seq + (size_t)s * BB * HH, nullptr, 0, BB, HH);
    }

    pack_enc_out<<<cdiv(BB * SS * 2 * HH, EW), EW, 0, stream>>>(hf_seq, hb_seq, enc_out);
    enc_cat_final<<<cdiv(BB * 2 * HH, EW), EW, 0, stream>>>(hf_seq, hb_seq, henc_cat);

    // hidden = tanh(henc_cat @ enc_fc_w^T + enc_fc_b)
    gemm_wmma_f32<1, 1><<<gemm_grid(BB, HD, 16, 1), 256, 0, stream>>>(
        henc_cat, 2 * HH, enc_fc_w, 2 * HH, enc_fc_b, hbuf0, HD, BB, HD, 2 * HH, 1, 2 * HH);

    // enc_proj = enc_out @ wa_enc^T + attn_b   (wa_enc = attn_w[:, 256:], row stride 768)
    gemm_wmma_f32<0, 2><<<gemm_grid(BB * SS, AA, 32, 1), 256, 0, stream>>>(
        enc_out, 2 * HH, attn_w + HD, HD + 2 * HH, attn_b, enc_proj, AA, BB * SS, AA, 2 * HH, 1, 2 * HH);

    // ---- decoder ----
    float* h_cur = hbuf0;
    float* h_nxt = hbuf1;
    const int KOUT = HD + 2 * HH + EE;  // 896
    for (int t = 0; t < TT - 1; ++t) {
        // attention + both concats (emb, w)
        attn_fused<<<BB, 256, 0, stream>>>(h_cur, attn_w, enc_proj, enc_out,
                                           trg, t, dec_emb, xcat, outcat);

        // xp = xcat @ dec_wih^T + dec_bih
        gemm_wmma_f32<0, 1><<<gemm_grid(BB, 3 * HD, 16, 1), 256, 0, stream>>>(
            xcat, EE + 2 * HH, dec_wih, EE + 2 * HH, dec_bih, xp_dec, 3 * HD,
            BB, 3 * HD, EE + 2 * HH, 1, EE + 2 * HH);
        // gh = h @ dec_whh^T + dec_bhh
        gemm_wmma_f32<0, 1><<<gemm_grid(BB, 3 * HD, 16, 1), 256, 0, stream>>>(
            h_cur, HD, dec_whh, HD, dec_bhh, gh_buf, 3 * HD, BB, 3 * HD, HD, 1, HD);

        // GRU gates; h_new also mirrored into outcat[:, 0:256]
        gru_elem<<<cdiv(BB * HD, EW), EW, 0, stream>>>(
            xp_dec, (long long)(3 * HD), gh_buf, h_cur, h_nxt, outcat, KOUT, BB, HD);

        // out[:, t+1, :] = outcat @ dec_fc_w^T + dec_fc_b  (split-K x4 for occupancy)
        gemm_wmma_f32<0, 1><<<gemm_grid(BB, VV, 16, KS_BIG), 256, 0, stream>>>(
            outcat, KOUT, dec_fc_w, KOUT, (const float*)nullptr,
            part, VV, BB, VV, KOUT, KS_BIG, KC_BIG);
        reduce_splitk<<<cdiv(BB * VV, EW), EW, 0, stream>>>(
            part, dec_fc_b, out + (size_t)(t + 1) * VV, (long long)TT * VV, BB, VV, KS_BIG);

        float* tmp = h_cur; h_cur = h_nxt; h_nxt = tmp;
    }
}